// YOLO_27522150433071
// MI455X (gfx1250) — compile-verified
//
#include <hip/hip_runtime.h>
#include <cstdint>
#include <cstddef>

// ---------------------------------------------------------------------------
// YOLO post-process for MI455X (gfx1250, wave32).
// Phase1 (128MB stream, ~5.5us @ 23.3TB/s) uses global_load_async_to_lds_b128.
// Pairwise-IoU area sums computed on the matrix pipe via v_wmma_f32_16x16x4_f32.
// IoU threshold evaluated division-free: iou>0.5  <=>  3*inter > sum + 1e-9.
// ---------------------------------------------------------------------------

constexpr int   BB    = 8;        // batch
constexpr int   NN    = 50000;    // anchors
constexpr int   CC    = 80;       // classes
constexpr int   KK    = 1000;     // top-k
constexpr int   KP    = 1024;     // padded k
constexpr int   RR    = 1008;     // padded rows for 16-row IoU tiles (63*16)
constexpr int   TILES = (NN + 31) / 32;
constexpr int   WPB   = 4;        // waves per block, phase 1
constexpr int   ADW   = 84;       // padded dwords per anchor in LDS (80 + pad)
constexpr float XCLIP = 4.135166556742356f;   // log(1000/16)

// ---- workspace layout (bytes) ---------------------------------------------
constexpr size_t OFF_SCORES = 0;
constexpr size_t OFF_CLASS  = OFF_SCORES + (size_t)BB * NN * 4;
constexpr size_t OFF_HIST   = OFF_CLASS  + (size_t)BB * NN * 4;
constexpr size_t OFF_CTR    = OFF_HIST   + (size_t)BB * 65536 * 4;
constexpr size_t OFF_SELS   = OFF_CTR    + (size_t)BB * 8 * 4;
constexpr size_t OFF_SELI   = OFF_SELS   + (size_t)BB * KP * 4;
constexpr size_t OFF_SRTS   = OFF_SELI   + (size_t)BB * KP * 4;
constexpr size_t OFF_SRTI   = OFF_SRTS   + (size_t)BB * KP * 4;
constexpr size_t OFF_BOX    = OFF_SRTI   + (size_t)BB * KP * 4;
constexpr size_t OFF_AREA   = OFF_BOX    + (size_t)BB * KP * 16;
constexpr size_t OFF_CLS2   = OFF_AREA   + (size_t)BB * KP * 4;
constexpr size_t OFF_MASK   = OFF_CLS2   + (size_t)BB * KP * 4;

typedef __attribute__((ext_vector_type(2))) float v2f;
typedef __attribute__((ext_vector_type(8))) float v8f;

#if defined(__HIP_DEVICE_COMPILE__) && __has_builtin(__builtin_amdgcn_wmma_f32_16x16x4_f32)
#define HAS_WMMA44 1
#else
#define HAS_WMMA44 0
#endif

__device__ __forceinline__ unsigned ballot32(bool p) {
#if defined(__HIP_DEVICE_COMPILE__) && __has_builtin(__builtin_amdgcn_ballot_w32)
  return __builtin_amdgcn_ballot_w32(p);
#else
  return (unsigned)__ballot(p);
#endif
}

// monotone order key: larger float -> larger unsigned
__device__ __forceinline__ unsigned okey(float s) {
  unsigned u = __float_as_uint(s);
  return (u & 0x80000000u) ? ~u : (u | 0x80000000u);
}

// ---------------------------------------------------------------------------
// Phase 1: per-anchor class max/argmax. Each wave stages a 32-anchor tile of
// logits (32 x 80 f32, padded to 84 dw/anchor) into LDS with async-to-LDS
// B128 copies, waits on ASYNCcnt, then each lane reduces its anchor from LDS.
// ---------------------------------------------------------------------------
__global__ void __launch_bounds__(WPB * 32)
k_phase1(const float* __restrict__ labels, float* __restrict__ scores,
         int* __restrict__ cls) {
  __shared__ float smem[WPB * 32 * ADW];
  const int wave = threadIdx.x >> 5;
  const int lane = threadIdx.x & 31;
  const int b    = blockIdx.y;
  const int tile = blockIdx.x * WPB + wave;
  if (tile >= TILES) return;
  const int aBase = tile * 32;
  const size_t imgBase = (size_t)b * NN * CC;
  float* wsm = &smem[wave * 32 * ADW];

  // 20 async B128 issues per wave: 32 anchors * 20 float4 each.
#pragma unroll
  for (int it = 0; it < 20; ++it) {
    int e   = it * 32 + lane;   // 0..639 (16B units within tile)
    int al  = e / 20;           // local anchor 0..31
    int sub = e % 20;           // float4 index within anchor
    int a   = aBase + al; if (a > NN - 1) a = NN - 1;   // clamp tail
    const float* gp = labels + imgBase + (size_t)a * CC + (size_t)sub * 4;
    unsigned lo = (unsigned)(size_t)(&wsm[al * ADW + sub * 4]);
    unsigned long long ga = (unsigned long long)(size_t)gp;
    asm volatile("global_load_async_to_lds_b128 %0, %1, off"
                 :: "v"(lo), "v"(ga) : "memory");
  }
  asm volatile("s_wait_asynccnt 0" ::: "memory");

  const int a = aBase + lane;
  float m = -3.402823466e38f;
  int am = 0;
  const float4* p4 = (const float4*)&wsm[lane * ADW];
#pragma unroll
  for (int i = 0; i < 20; ++i) {
    float4 v = p4[i];
    if (v.x > m) { m = v.x; am = 4 * i + 0; }
    if (v.y > m) { m = v.y; am = 4 * i + 1; }
    if (v.z > m) { m = v.z; am = 4 * i + 2; }
    if (v.w > m) { m = v.w; am = 4 * i + 3; }
  }
  if (a < NN) {
    float prob = 1.0f / (1.0f + __expf(-m));   // sigmoid(max) == max(sigmoid)
    bool fg = (am != 0) && (prob > 0.05f);
    scores[(size_t)b * NN + a] = fg ? prob : -1.0f;
    cls[(size_t)b * NN + a]    = am;
  }
}

// ---------------------------------------------------------------------------
// Radix-select (two 16-bit passes) for exact top-K threshold key.
// ---------------------------------------------------------------------------
__global__ void k_zero(unsigned* p, int n) {
  int i = blockIdx.x * blockDim.x + threadIdx.x;
  if (i < n) p[i] = 0u;
}

__global__ void k_hist1(const float* __restrict__ scores, unsigned* __restrict__ hist) {
  int b = blockIdx.y;
  int i0 = blockIdx.x * blockDim.x + threadIdx.x;
  int st = gridDim.x * blockDim.x;
  for (int n = i0; n < NN; n += st) {
    unsigned u = okey(scores[(size_t)b * NN + n]);
    atomicAdd(&hist[(size_t)b * 65536 + (u >> 16)], 1u);
  }
}

__global__ void k_findthr1(const unsigned* __restrict__ hist, int* __restrict__ ctr) {
  int b = blockIdx.x;
  __shared__ unsigned cs[256];
  const unsigned* h = hist + (size_t)b * 65536;
  unsigned s = 0;
  for (int i = 0; i < 256; ++i) s += h[threadIdx.x * 256 + i];
  cs[threadIdx.x] = s;
  __syncthreads();
  if (threadIdx.x == 0) {
    int acc = 0, T1 = -1, C1 = 0; bool found = false;
    for (int t = 255; t >= 0; --t) {
      if (acc + (int)cs[t] >= KK) {
        for (int i = 255; i >= 0; --i) {
          int c = (int)h[t * 256 + i];
          if (acc + c >= KK) { T1 = t * 256 + i; C1 = acc; found = true; break; }
          acc += c;
        }
        break;
      }
      acc += (int)cs[t];
    }
    if (!found) { T1 = -1; C1 = acc; }
    ctr[b * 8 + 2] = T1;
    ctr[b * 8 + 3] = C1;
  }
}

__global__ void k_hist2(const float* __restrict__ scores, const int* __restrict__ ctr,
                        unsigned* __restrict__ hist) {
  int b = blockIdx.y;
  int T1 = ctr[b * 8 + 2];
  int i0 = blockIdx.x * blockDim.x + threadIdx.x;
  int st = gridDim.x * blockDim.x;
  for (int n = i0; n < NN; n += st) {
    unsigned u = okey(scores[(size_t)b * NN + n]);
    if ((int)(u >> 16) == T1)
      atomicAdd(&hist[(size_t)b * 65536 + (u & 0xFFFFu)], 1u);
  }
}

__global__ void k_findthr2(const unsigned* __restrict__ hist, int* __restrict__ ctr,
                           float* __restrict__ selS, int* __restrict__ selI) {
  int b = blockIdx.x;
  __shared__ unsigned cs[256];
  const unsigned* h = hist + (size_t)b * 65536;
  unsigned s = 0;
  for (int i = 0; i < 256; ++i) s += h[threadIdx.x * 256 + i];
  cs[threadIdx.x] = s;
  __syncthreads();
  if (threadIdx.x == 0) {
    int T1 = ctr[b * 8 + 2], C1 = ctr[b * 8 + 3];
    unsigned thrKey = 0u; int nAbove = C1;
    if (T1 >= 0) {
      int rem = KK - C1;
      int acc = 0, T2 = 0; bool fnd = false;
      for (int t = 255; t >= 0 && !fnd; --t) {
        if (acc + (int)cs[t] >= rem) {
          for (int i = 255; i >= 0; --i) {
            int c = (int)h[t * 256 + i];
            if (acc + c >= rem) { T2 = t * 256 + i; fnd = true; break; }
            acc += c;
          }
        } else acc += (int)cs[t];
      }
      thrKey = ((unsigned)T1 << 16) | (unsigned)T2;
      nAbove = C1 + acc;               // strictly above exact key
    }
    ctr[b * 8 + 0] = 0;                // counter for strictly-above slots
    ctr[b * 8 + 1] = nAbove;           // counter base for tie slots
    ctr[b * 8 + 4] = (int)thrKey;
  }
  for (int k = threadIdx.x; k < KP; k += blockDim.x) {
    selS[b * KP + k] = -3.402823466e38f;
    selI[b * KP + k] = 0;
  }
}

__global__ void k_select(const float* __restrict__ scores, int* __restrict__ ctr,
                         float* __restrict__ selS, int* __restrict__ selI) {
  int b = blockIdx.y;
  unsigned thrKey = (unsigned)ctr[b * 8 + 4];
  int i0 = blockIdx.x * blockDim.x + threadIdx.x;
  int st = gridDim.x * blockDim.x;
  for (int n = i0; n < NN; n += st) {
    float s = scores[(size_t)b * NN + n];
    unsigned u = okey(s);
    if (u > thrKey) {
      int p = atomicAdd(&ctr[b * 8 + 0], 1);
      if (p < KK) { selS[b * KP + p] = s; selI[b * KP + p] = n; }
    } else if (u == thrKey) {
      int p = atomicAdd(&ctr[b * 8 + 1], 1);
      if (p < KK) { selS[b * KP + p] = s; selI[b * KP + p] = n; }
    }
  }
}

// exact descending order (ties by anchor index) via rank-by-counting
__global__ void __launch_bounds__(KP)
k_sort(const float* __restrict__ selS, const int* __restrict__ selI,
       float* __restrict__ srtS, int* __restrict__ srtI) {
  __shared__ float ss[KP];
  __shared__ int   si[KP];
  int b = blockIdx.x, t = threadIdx.x;
  float s = selS[b * KP + t];
  int   id = selI[b * KP + t];
  ss[t] = s; si[t] = id;
  __syncthreads();
  int r = 0;
  for (int j = 0; j < KP; ++j) {
    float sj = ss[j]; int ij = si[j];
    r += (sj > s) || (sj == s && (ij < id || (ij == id && j < t)));
  }
  srtS[b * KP + r] = s;
  srtI[b * KP + r] = id;
}

__global__ void k_decode(const int* __restrict__ srtI, const float* __restrict__ codes,
                         const float* __restrict__ anch, const int* __restrict__ cls,
                         float* __restrict__ boxes, float* __restrict__ areas,
                         int* __restrict__ cls2) {
  int b = blockIdx.y;
  int k = blockIdx.x * blockDim.x + threadIdx.x;
  if (k >= KP) return;
  float4 box = make_float4(0.f, 0.f, 0.f, 0.f);
  float ar = 0.f; int c = 0;
  if (k < KK) {
    int idx = srtI[b * KP + k];
    const float* cd = codes + ((size_t)b * NN + idx) * 4;
    const float* an = anch  + ((size_t)b * NN + idx) * 4;
    float w  = an[2] - an[0], h = an[3] - an[1];
    float cx = an[0] + 0.5f * w, cy = an[1] + 0.5f * h;
    float dw = fminf(cd[2], XCLIP), dh = fminf(cd[3], XCLIP);
    float pcx = cd[0] * w + cx, pcy = cd[1] * h + cy;
    float pw = __expf(dw) * w, ph = __expf(dh) * h;
    box = make_float4(pcx - 0.5f * pw, pcy - 0.5f * ph, pcx + 0.5f * pw, pcy + 0.5f * ph);
    ar = (box.z - box.x) * (box.w - box.y);
    c  = cls[(size_t)b * NN + idx];
  }
  ((float4*)boxes)[(size_t)b * KP + k] = box;
  areas[(size_t)b * KP + k] = ar;
  cls2[(size_t)b * KP + k]  = c;
}

// ---------------------------------------------------------------------------
// IoU suppression bitmask. One wave per 16x32 tile of pairs. Area sums
// (area_i + area_j) for each 16x16 half come from v_wmma_f32_16x16x4_f32 as a
// rank-2 update: A = [rowArea | 1], B = [1 ; colArea]. The IoU threshold is
// evaluated without division (denominator is strictly positive):
//   inter/(sum-inter+eps) > 0.5  <=>  3*inter > sum + eps.
// Predicates packed via wave32 ballots into one u32 mask word per row.
// ---------------------------------------------------------------------------
__global__ void __launch_bounds__(32)
k_iou(const float* __restrict__ boxes, const float* __restrict__ areas,
      unsigned* __restrict__ mask) {
  __shared__ float4  rbox[16];
  __shared__ float   rA[16];
  __shared__ unsigned bal[16];
  const int word = blockIdx.x;   // 0..31  (cols word*32 .. +31)
  const int tR   = blockIdx.y;   // 0..62  (rows tR*16 .. +15)
  const int b    = blockIdx.z;
  const int lane = threadIdx.x;
  const int i0   = tR * 16;
  const float4* bx = (const float4*)boxes + (size_t)b * KP;
  const float*  ar = areas + (size_t)b * KP;
  if (lane < 16) { rbox[lane] = bx[i0 + lane]; rA[lane] = ar[i0 + lane]; }
  __syncthreads();
  const int jbase = word * 32;
  const int hi8   = (lane >> 4) << 3;   // row offset for D layout (+8 in lanes 16..31)
  const bool lo16 = lane < 16;
  const float rAl = rA[lane & 15];

#pragma unroll
  for (int h = 0; h < 2; ++h) {
    const int jc = jbase + h * 16 + (lane & 15);
    const float4 cb = bx[jc];
    const float  ca = ar[jc];
#if HAS_WMMA44
    // A 16x4 f32: VGPR0 = K0 (lanes 0-15) / K2 (lanes 16-31); VGPR1 = K1/K3.
    v2f a, bm;
    a.x  = lo16 ? rAl  : 0.0f;   // K0 column of A = row areas
    a.y  = lo16 ? 1.0f : 0.0f;   // K1 column of A = ones
    bm.x = lo16 ? 1.0f : 0.0f;   // K0 row of B = ones
    bm.y = lo16 ? ca   : 0.0f;   // K1 row of B = col areas
    v8f cacc = {0.f, 0.f, 0.f, 0.f, 0.f, 0.f, 0.f, 0.f};
    v8f d = __builtin_amdgcn_wmma_f32_16x16x4_f32(
        false, a, false, bm, (short)0, cacc, false, false);
#endif
#pragma unroll
    for (int k = 0; k < 8; ++k) {
      float4 rb = rbox[(k + hi8) & 15];
#if HAS_WMMA44
      float sum = d[k];                    // area_row + area_col from matrix pipe
#else
      float sum = rA[(k + hi8) & 15] + ca; // VALU fallback
#endif
      float ix1 = fmaxf(rb.x, cb.x), iy1 = fmaxf(rb.y, cb.y);
      float ix2 = fminf(rb.z, cb.z), iy2 = fminf(rb.w, cb.w);
      float iw = fmaxf(ix2 - ix1, 0.f), ih = fmaxf(iy2 - iy1, 0.f);
      float inter = iw * ih;
      // division-free threshold: iou > 0.5  <=>  3*inter > sum + 1e-9
      unsigned bl = ballot32(3.0f * inter > sum + 1e-9f);
      if (lane == 0) bal[h * 8 + k] = bl;
    }
  }
  __syncthreads();
  if (lane < 16) {
    unsigned lo = bal[lane & 7];        // col half 0 (word bits 0..15)
    unsigned hi = bal[8 + (lane & 7)];  // col half 1 (word bits 16..31)
    unsigned v = (lane < 8)
        ? ((lo & 0xFFFFu) | ((hi & 0xFFFFu) << 16))
        : ((lo >> 16) | ((hi >> 16) << 16));
    mask[(((size_t)b * RR) + (i0 + lane)) * 32 + word] = v;
  }
}

// ---------------------------------------------------------------------------
// Sequential greedy NMS: one wave per image; lane w owns keep word w.
// ---------------------------------------------------------------------------
__global__ void __launch_bounds__(32)
k_nms(const unsigned* __restrict__ mask, const float* __restrict__ srtS,
      const float* __restrict__ boxes, const int* __restrict__ cls2,
      float* __restrict__ out) {
  const int b = blockIdx.x, lane = threadIdx.x;
  unsigned keepw = 0xFFFFFFFFu;
  const unsigned* mrow = mask + (size_t)b * RR * 32;
  for (int i = 0; i < KK; ++i) {
    unsigned m = mrow[i * 32 + lane];                    // unconditional: prefetchable
    unsigned kwi = __shfl(keepw, i >> 5, 32);            // wave32 broadcast
    if ((kwi >> (i & 31)) & 1u) {                        // uniform branch
      int wo = i >> 5;
      unsigned g = (lane < wo) ? 0u
                 : (lane > wo) ? 0xFFFFFFFFu
                 : (((i & 31) == 31) ? 0u : (0xFFFFFFFFu << ((i & 31) + 1)));
      keepw &= ~(m & g);                                 // suppress j > i
    }
  }
  __shared__ unsigned kws[32];
  kws[lane] = keepw;
  __syncthreads();
  const float*  sS  = srtS + (size_t)b * KP;
  const float4* bxp = (const float4*)boxes + (size_t)b * KP;
  const int*    cl  = cls2 + (size_t)b * KP;
  float* oCls  = out;
  float* oScr  = out + (size_t)BB * KK;
  float* oBox  = out + (size_t)2 * BB * KK;
  float* oKeep = out + (size_t)6 * BB * KK;
  for (int k = lane; k < KK; k += 32) {
    float s = sS[k];
    bool kp = (((kws[k >> 5] >> (k & 31)) & 1u) != 0u) && (s > 0.0f);
    float4 bv = bxp[k];
    oCls[b * KK + k] = kp ? (float)cl[k] : -1.0f;
    oScr[b * KK + k] = kp ? s : 0.0f;
    ((float4*)oBox)[b * KK + k] = kp ? bv : make_float4(0.f, 0.f, 0.f, 0.f);
    oKeep[b * KK + k] = kp ? 1.0f : 0.0f;
  }
}

// ---------------------------------------------------------------------------
extern "C" void kernel_launch(void* const* d_in, const int* in_sizes, int n_in,
                              void* d_out, int out_size, void* d_ws, size_t ws_size,
                              hipStream_t stream) {
  (void)in_sizes; (void)n_in; (void)out_size; (void)ws_size;
  const float* labels = (const float*)d_in[0];   // (B,N,C)
  const float* codes  = (const float*)d_in[1];   // (B,N,4)
  const float* anch   = (const float*)d_in[2];   // (B,N,4)
  float* out = (float*)d_out;
  char* ws = (char*)d_ws;

  float*    scores = (float*)(ws + OFF_SCORES);
  int*      cls    = (int*)(ws + OFF_CLASS);
  unsigned* hist   = (unsigned*)(ws + OFF_HIST);
  int*      ctr    = (int*)(ws + OFF_CTR);
  float*    selS   = (float*)(ws + OFF_SELS);
  int*      selI   = (int*)(ws + OFF_SELI);
  float*    srtS   = (float*)(ws + OFF_SRTS);
  int*      srtI   = (int*)(ws + OFF_SRTI);
  float*    boxes  = (float*)(ws + OFF_BOX);
  float*    areas  = (float*)(ws + OFF_AREA);
  int*      cls2   = (int*)(ws + OFF_CLS2);
  unsigned* maskp  = (unsigned*)(ws + OFF_MASK);

  const int histN = BB * 65536;
  k_zero<<<(histN + 255) / 256, 256, 0, stream>>>(hist, histN);
  k_phase1<<<dim3((TILES + WPB - 1) / WPB, BB), WPB * 32, 0, stream>>>(labels, scores, cls);
  k_hist1<<<dim3(64, BB), 256, 0, stream>>>(scores, hist);
  k_findthr1<<<BB, 256, 0, stream>>>(hist, ctr);
  k_zero<<<(histN + 255) / 256, 256, 0, stream>>>(hist, histN);
  k_hist2<<<dim3(64, BB), 256, 0, stream>>>(scores, ctr, hist);
  k_findthr2<<<BB, 256, 0, stream>>>(hist, ctr, selS, selI);
  k_select<<<dim3(64, BB), 256, 0, stream>>>(scores, ctr, selS, selI);
  k_sort<<<BB, KP, 0, stream>>>(selS, selI, srtS, srtI);
  k_decode<<<dim3(KP / 256, BB), 256, 0, stream>>>(srtI, codes, anch, cls, boxes, areas, cls2);
  k_iou<<<dim3(32, RR / 16, BB), 32, 0, stream>>>(boxes, areas, maskp);
  k_nms<<<BB, 32, 0, stream>>>(maskp, srtS, boxes, cls2, out);
}